// Channel_Attn_46746424050494
// MI455X (gfx1250) — compile-verified
//
#include <hip/hip_runtime.h>

typedef float v2f __attribute__((ext_vector_type(2)));
typedef float v4f __attribute__((ext_vector_type(4)));
typedef float v8f __attribute__((ext_vector_type(8)));

#define Bn   32
#define Cn   512
#define HW2  1024
#define NW   77
#define NP   80     // n padded to multiple of 16
#define WD   256

static __device__ __forceinline__ v8f wmma_f32(v2f a, v2f b, v8f c) {
  return __builtin_amdgcn_wmma_f32_16x16x4_f32(false, a, false, b, (short)0, c, false, false);
}

// ---------------------------------------------------------------------------
// Kernel 1: we[b, n(0..79), k] = word_emb[b] @ W_fc^T + b_fc, written twice:
//   we_ws  : (b, n, k)  row-major   -> contiguous-in-k B loads for kernel 2
//   weT_ws : (b, k, n)  row-major   -> contiguous-in-n B loads for kernel 3
// Padded rows n=77..79 duplicate row 76: never observed downstream (scores
// cols >=77 masked to -inf; att cols >=77 stored as exact 0), so no masking
// is needed in the K-loop.
// One wave = one 16x16 tile. Tiles/batch: 5 (n) x 64 (k). 8 waves / block.
// ---------------------------------------------------------------------------
__global__ __launch_bounds__(256) void we_gemm_kernel(
    const float* __restrict__ word_emb, const float* __restrict__ W_fc,
    const float* __restrict__ b_fc, float* __restrict__ we_ws,
    float* __restrict__ weT_ws) {
  const int lane = threadIdx.x & 31;
  const int wave = threadIdx.x >> 5;
  int tile = blockIdx.x * 8 + wave;          // 0 .. Bn*5*64-1
  const int b  = tile / (5 * 64);
  tile -= b * 5 * 64;
  const int tm = tile >> 6;                  // n-tile 0..4
  const int tn = tile & 63;                  // k-tile 0..63
  const int half = lane >> 4;
  const int l16  = lane & 15;

  const int a_row = tm * 16 + l16;           // word index (clamped if >= 77)
  const int col   = tn * 16 + l16;           // output k column
  const float* Abase = word_emb + ((size_t)b * NW + (a_row < NW ? a_row : NW - 1)) * WD;
  const float* Bbase = W_fc + (size_t)col * WD;   // W_fc is (1024,256) row-major

  v8f acc = {};
  for (int k0 = 0; k0 < WD; k0 += 4) {
    const int k = k0 + 2 * half;             // this lane holds K = k, k+1
    v2f a  = *(const v2f*)(Abase + k);
    v2f bm = *(const v2f*)(Bbase + k);
    acc = wmma_f32(a, bm, acc);
  }
  const float bias = b_fc[col];

  // (n, k) layout: 8 stores, stride HW2 rows
  float* D = we_ws + ((size_t)b * NP + tm * 16 + 8 * half) * HW2 + col;
  #pragma unroll
  for (int i = 0; i < 8; ++i) D[(size_t)i * HW2] = acc[i] + bias;

  // (k, n) layout: 8 contiguous floats per lane -> two b128 stores
  float* T = weT_ws + ((size_t)b * HW2 + col) * NP + tm * 16 + 8 * half;
  v4f lo = {acc[0] + bias, acc[1] + bias, acc[2] + bias, acc[3] + bias};
  v4f hi = {acc[4] + bias, acc[5] + bias, acc[6] + bias, acc[7] + bias};
  *(v4f*)(T)     = lo;
  *(v4f*)(T + 4) = hi;
}

// ---------------------------------------------------------------------------
// Kernel 2: scores = f @ we^T  (K = 1024), fused softmax over n, write att.
// One wave owns a 16-row c-slab x all 80 n-columns (5 accumulators; one A
// fragment feeds 5 WMMAs per K-step).
// ---------------------------------------------------------------------------
__global__ __launch_bounds__(256) void scores_softmax_kernel(
    const float* __restrict__ feat, const float* __restrict__ we_ws,
    float* __restrict__ att_ws) {
  const int lane = threadIdx.x & 31;
  const int wave = threadIdx.x >> 5;
  const int slab = blockIdx.x * 8 + wave;    // 0 .. Bn*32-1
  const int b  = slab >> 5;
  const int cs = (slab & 31) * 16;
  const int half = lane >> 4;
  const int l16  = lane & 15;

  const float* Abase  = feat  + ((size_t)b * Cn + cs + l16) * HW2;
  const float* Bbase0 = we_ws + ((size_t)b * NP + l16) * HW2;   // + 16t rows

  v8f acc[5] = {};
  for (int k0 = 0; k0 < HW2; k0 += 4) {
    const int k = k0 + 2 * half;
    v2f a = *(const v2f*)(Abase + k);
    #pragma unroll
    for (int t = 0; t < 5; ++t) {
      v2f bm = *(const v2f*)(Bbase0 + (size_t)(16 * t) * HW2 + k);
      acc[t] = wmma_f32(a, bm, acc[t]);
    }
  }

  // softmax over n (80 cols, mask >= 77). Row r = cs + i + 8*half lives in the
  // 16 lanes of this half: xor masks 8..1 stay inside the half-group.
  float* Dbase = att_ws + ((size_t)b * Cn + cs + 8 * half) * NP;
  #pragma unroll
  for (int i = 0; i < 8; ++i) {
    float m = -3.402823466e38f;
    #pragma unroll
    for (int t = 0; t < 5; ++t)
      if (16 * t + l16 < NW) m = fmaxf(m, acc[t][i]);
    #pragma unroll
    for (int off = 8; off >= 1; off >>= 1) m = fmaxf(m, __shfl_xor(m, off, 32));

    float e[5], s = 0.f;
    #pragma unroll
    for (int t = 0; t < 5; ++t) {
      e[t] = (16 * t + l16 < NW) ? __expf(acc[t][i] - m) : 0.f;
      s += e[t];
    }
    #pragma unroll
    for (int off = 8; off >= 1; off >>= 1) s += __shfl_xor(s, off, 32);
    const float inv = 1.f / s;
    #pragma unroll
    for (int t = 0; t < 5; ++t)
      Dbase[(size_t)i * NP + 16 * t + l16] = e[t] * inv;   // exact 0 for n>=77
  }
}

// ---------------------------------------------------------------------------
// Kernel 3: att_feat = att @ we  (K = 80). One wave = one 16x64 out tile
// (4 accumulators: A fragment reused 4x; B from weT -> contiguous b64 loads).
// att cols 77..79 are exact 0, so padded-K terms vanish.
// ---------------------------------------------------------------------------
__global__ __launch_bounds__(256) void attfeat_kernel(
    const float* __restrict__ att_ws, const float* __restrict__ weT_ws,
    float* __restrict__ out) {
  const int lane = threadIdx.x & 31;
  const int wave = threadIdx.x >> 5;
  int tile = blockIdx.x * 8 + wave;          // 0 .. Bn*32*16-1
  const int b = tile / (32 * 16);
  tile -= b * 32 * 16;
  const int tm = tile >> 4;                  // c-tile 0..31
  const int tq = tile & 15;                  // 64-wide n-quad 0..15
  const int half = lane >> 4;
  const int l16  = lane & 15;

  const float* Abase = att_ws + ((size_t)b * Cn + tm * 16 + l16) * NP;
  // B[k][ncol] = weT[(b*HW2 + ncol)*NP + k], ncol = tq*64 + 16*t + l16
  const float* Bbase0 = weT_ws + ((size_t)b * HW2 + tq * 64 + l16) * NP;

  v8f acc[4] = {};
  for (int k0 = 0; k0 < NP; k0 += 4) {
    const int k = k0 + 2 * half;
    v2f a = *(const v2f*)(Abase + k);
    #pragma unroll
    for (int t = 0; t < 4; ++t) {
      v2f bm = *(const v2f*)(Bbase0 + (size_t)(16 * t) * NP + k);
      acc[t] = wmma_f32(a, bm, acc[t]);
    }
  }
  float* D = out + ((size_t)b * Cn + tm * 16 + 8 * half) * HW2 + tq * 64 + l16;
  #pragma unroll
  for (int t = 0; t < 4; ++t)
    #pragma unroll
    for (int i = 0; i < 8; ++i)
      D[(size_t)i * HW2 + 16 * t] = acc[t][i];
}

// ---------------------------------------------------------------------------
extern "C" void kernel_launch(void* const* d_in, const int* in_sizes, int n_in,
                              void* d_out, int out_size, void* d_ws, size_t ws_size,
                              hipStream_t stream) {
  const float* feat     = (const float*)d_in[0];
  const float* word_emb = (const float*)d_in[1];
  const float* W_fc     = (const float*)d_in[2];
  const float* b_fc     = (const float*)d_in[3];
  float* out = (float*)d_out;

  float* we_ws  = (float*)d_ws;                        // Bn*NP*HW2  = 10.5 MB
  float* weT_ws = we_ws  + (size_t)Bn * NP * HW2;      // Bn*HW2*NP  = 10.5 MB
  float* att_ws = weT_ws + (size_t)Bn * HW2 * NP;      // Bn*Cn*NP   =  5.2 MB

  we_gemm_kernel       <<<(Bn * 5 * 64) / 8,  256, 0, stream>>>(word_emb, W_fc, b_fc, we_ws, weT_ws);
  scores_softmax_kernel<<<(Bn * 32) / 8,      256, 0, stream>>>(feat, we_ws, att_ws);
  attfeat_kernel       <<<(Bn * 32 * 16) / 8, 256, 0, stream>>>(att_ws, weT_ws, out);
}